// MergedEmbeddingBag_84859963834386
// MI455X (gfx1250) — compile-verified
//
#include <hip/hip_runtime.h>

// Merged embedding-bag, sum pooling. 26 tables, DIM=128, 214 packed index
// columns per sample, B=32768. Memory-bound gather:
//  - "Big" tables 0..4 (1M/400k/200k/100k/100k rows, ~0.9GB total) are the
//    only true HBM misses. One wave per sample fuses all five: 14 nt-hinted
//    global_load_b128 issued back-to-back -> 14 concurrent HBM misses/wave.
//  - Small tables 5..25 are L2-resident (192MB L2). One wave per (sample,
//    table), hot counts folded in at compile time and fully unrolled.
// Indices are fetched via the scalar (SMEM) path; output is streamed with
// non-temporal stores (written once, never re-read).

typedef float v4f __attribute__((ext_vector_type(4)));

#define NTAB 26
#define DIM 128
#define IDX_COLS 214
#define OUT_STRIDE (NTAB * DIM) // 3328
#define NSMALL 21               // tables 5..25
#define BIG_HOTS 14             // 3+2+1+2+6

template <bool STREAM>
__device__ __forceinline__ v4f load_row(const float* lane_base, int row) {
  const v4f* p = (const v4f*)(lane_base + (size_t)row * DIM);
  if (STREAM) return __builtin_nontemporal_load(p);
  return *p;
}

// Compile-time hot count: fully unrolled (all loads in flight before the
// first wait); large counts chunked by 8 to bound VGPR pressure (those
// tables are L2-resident, 8-deep covers L2 latency).
template <int HOTS, bool STREAM>
__device__ __forceinline__ v4f pool_fixed(const int* __restrict__ ip,
                                          const float* __restrict__ lane_base) {
  v4f acc = {0.0f, 0.0f, 0.0f, 0.0f};
  if constexpr (HOTS <= 16) {
#pragma unroll
    for (int j = 0; j < HOTS; ++j) acc += load_row<STREAM>(lane_base, ip[j]);
  } else {
    constexpr int CH = 8;
#pragma unroll 1
    for (int c = 0; c < HOTS / CH; ++c) {
#pragma unroll
      for (int k = 0; k < CH; ++k) acc += load_row<STREAM>(lane_base, ip[k]);
      ip += CH;
    }
#pragma unroll
    for (int j = 0; j < HOTS % CH; ++j) acc += load_row<STREAM>(lane_base, ip[j]);
  }
  return acc;
}

__global__ __launch_bounds__(256)
void embbag_kernel(const int* __restrict__ idx, const float* __restrict__ w,
                   float* __restrict__ out, int B) {
  const int lane  = (int)(threadIdx.x & 31u);
  // Wave slot into an SGPR so per-wave ids / index pointers are uniform ->
  // index fetches take the scalar (s_load) path, off the vector pipe.
  const int wslot = __builtin_amdgcn_readfirstlane((int)(threadIdx.x >> 5));
  const int wid   = (int)blockIdx.x * 8 + wslot;
  if (wid >= B * (1 + NSMALL)) return;

  const float* lb = w + lane * 4; // fold the lane offset once

  if (wid < B) {
    // ---- Big wave: tables 0..4 of sample b, 14 HBM misses in flight ----
    const int b = wid;
    const int* ip = idx + (size_t)b * IDX_COLS; // hstart(0..4)=0,3,5,6,8
    v4f v[BIG_HOTS];
#pragma unroll
    for (int j = 0; j < BIG_HOTS; ++j) v[j] = load_row<true>(lb, ip[j]);

    v4f a0 = (v[0] + v[1]) + v[2];                                // t0, hot 3
    v4f a1 = v[3] + v[4];                                         // t1, hot 2
    v4f a2 = v[5];                                                // t2, hot 1
    v4f a3 = v[6] + v[7];                                         // t3, hot 2
    v4f a4 = ((((v[8] + v[9]) + v[10]) + v[11]) + v[12]) + v[13]; // t4, hot 6

    float* ob = out + (size_t)b * OUT_STRIDE + lane * 4;
    __builtin_nontemporal_store(a0, (v4f*)(ob + 0 * DIM));
    __builtin_nontemporal_store(a1, (v4f*)(ob + 1 * DIM));
    __builtin_nontemporal_store(a2, (v4f*)(ob + 2 * DIM));
    __builtin_nontemporal_store(a3, (v4f*)(ob + 3 * DIM));
    __builtin_nontemporal_store(a4, (v4f*)(ob + 4 * DIM));
    return;
  }

  // ---- Small wave: one (sample, table) segment, table t = r + 5 ----
  const int wid2 = wid - B;
  const int b = wid2 / NSMALL;
  const int r = wid2 - b * NSMALL;

  const int* ip0 = idx + (size_t)b * IDX_COLS;
  // HOTS(t=5..25)   = {1,1,1,1,7,3,8,1,6,9,5,1,1,1,12,100,27,10,3,1,1}
  // HSTART(t=5..25) = {14,15,16,17,18,25,28,36,37,43,52,57,58,59,60,72,
  //                    172,199,209,212,213} folded in as immediates.
  v4f acc;
  switch (r) {
    case  0: acc = pool_fixed<  1, false>(ip0 +  14, lb); break; // t=5
    case  1: acc = pool_fixed<  1, false>(ip0 +  15, lb); break; // t=6
    case  2: acc = pool_fixed<  1, false>(ip0 +  16, lb); break; // t=7
    case  3: acc = pool_fixed<  1, false>(ip0 +  17, lb); break; // t=8
    case  4: acc = pool_fixed<  7, false>(ip0 +  18, lb); break; // t=9
    case  5: acc = pool_fixed<  3, false>(ip0 +  25, lb); break; // t=10
    case  6: acc = pool_fixed<  8, false>(ip0 +  28, lb); break; // t=11
    case  7: acc = pool_fixed<  1, false>(ip0 +  36, lb); break; // t=12
    case  8: acc = pool_fixed<  6, false>(ip0 +  37, lb); break; // t=13
    case  9: acc = pool_fixed<  9, false>(ip0 +  43, lb); break; // t=14
    case 10: acc = pool_fixed<  5, false>(ip0 +  52, lb); break; // t=15
    case 11: acc = pool_fixed<  1, false>(ip0 +  57, lb); break; // t=16
    case 12: acc = pool_fixed<  1, false>(ip0 +  58, lb); break; // t=17
    case 13: acc = pool_fixed<  1, false>(ip0 +  59, lb); break; // t=18
    case 14: acc = pool_fixed< 12, false>(ip0 +  60, lb); break; // t=19
    case 15: acc = pool_fixed<100, false>(ip0 +  72, lb); break; // t=20
    case 16: acc = pool_fixed< 27, false>(ip0 + 172, lb); break; // t=21
    case 17: acc = pool_fixed< 10, false>(ip0 + 199, lb); break; // t=22
    case 18: acc = pool_fixed<  3, false>(ip0 + 209, lb); break; // t=23
    case 19: acc = pool_fixed<  1, false>(ip0 + 212, lb); break; // t=24
    case 20: acc = pool_fixed<  1, false>(ip0 + 213, lb); break; // t=25
    default: __builtin_unreachable();
  }

  const int t = r + 5;
  v4f* op = (v4f*)(out + (size_t)b * OUT_STRIDE + t * DIM + lane * 4);
  __builtin_nontemporal_store(acc, op);
}

extern "C" void kernel_launch(void* const* d_in, const int* in_sizes, int n_in,
                              void* d_out, int out_size, void* d_ws, size_t ws_size,
                              hipStream_t stream) {
  const int*   idx = (const int*)d_in[0];   // [B, 214] int32, absolute rows
  const float* w   = (const float*)d_in[1]; // [TOTAL_ROWS, 128] fp32
  float*       out = (float*)d_out;         // [B, 3328] fp32

  const int B = in_sizes[0] / IDX_COLS;
  const int nwaves = B * (1 + NSMALL);      // B big waves + B*21 small waves
  const int waves_per_block = 256 / 32;
  const int blocks = (nwaves + waves_per_block - 1) / waves_per_block;

  embbag_kernel<<<blocks, 256, 0, stream>>>(idx, w, out, B);
}